// NearestNeighbor_26242250179143
// MI455X (gfx1250) — compile-verified
//
#include <hip/hip_runtime.h>

typedef float v2f __attribute__((ext_vector_type(2)));
typedef float v8f __attribute__((ext_vector_type(8)));

#define DOBS 1056   // T_OBS * D = 16 * 66 floats per sample

// ---------------------------------------------------------------------------
// Kernel 1: initialize the global argmin key to +inf
// ---------------------------------------------------------------------------
__global__ void nn_init_kernel(unsigned long long* ws) {
    if (threadIdx.x == 0 && blockIdx.x == 0) ws[0] = ~0ull;
}

// ---------------------------------------------------------------------------
// Kernel 2: streaming SSE + argmin.
// Each wave handles 16 samples per group via V_WMMA_F32_16X16X4_F32 with an
// all-ones B matrix: D[i][*] = sum_k A[i][k], so the accumulator holds the
// per-sample sum of squared diffs directly (no shuffle reduction needed).
// Because B == 1 everywhere, the mapping of data elements onto (step, K-slot)
// is arbitrary -> each lane streams a contiguous half-row with float4 loads.
// ---------------------------------------------------------------------------
__global__ __launch_bounds__(256) void nn_dist_kernel(
    const float* __restrict__ q,      // [1056] query
    const float* __restrict__ obs,    // [N, 1056]
    int n, int n_groups,
    unsigned long long* __restrict__ ws) {

    __shared__ float sq[DOBS];
    for (int i = threadIdx.x; i < DOBS; i += blockDim.x) sq[i] = q[i];
    __syncthreads();

    const int lane   = threadIdx.x & 31;
    const int row    = lane & 15;          // sample row within the 16-group
    const int half   = lane >> 4;          // 0 -> cols [0,528), 1 -> [528,1056)
    const int colOff = half * (DOBS / 2);

    const int waveId = (int)((blockIdx.x * blockDim.x + threadIdx.x) >> 5);
    const int nWaves = (int)((gridDim.x * blockDim.x) >> 5);

    v2f bOnes; bOnes.x = 1.0f; bOnes.y = 1.0f;   // all-ones B, layout-free
    unsigned long long bestKey = ~0ull;

    const float4* qv = (const float4*)(sq + colOff);

    for (int g = waveId; g < n_groups; g += nWaves) {
        int s  = g * 16 + row;
        int sc = (s < n) ? s : (n - 1);    // clamp tail rows (masked later)
        const float4* xv = (const float4*)(obs + (size_t)sc * DOBS + colOff);

        v8f acc = {};
        #pragma unroll 4
        for (int j = 0; j < DOBS / 8; ++j) {       // 132 float4 per lane
            float4 x  = xv[j];
            float4 qq = qv[j];
            float d0 = x.x - qq.x, d1 = x.y - qq.y;
            float d2 = x.z - qq.z, d3 = x.w - qq.w;
            v2f a0; a0.x = d0 * d0; a0.y = d1 * d1;
            v2f a1; a1.x = d2 * d2; a1.y = d3 * d3;
            acc = __builtin_amdgcn_wmma_f32_16x16x4_f32(
                      false, a0, false, bOnes, (short)0, acc, false, false);
            acc = __builtin_amdgcn_wmma_f32_16x16x4_f32(
                      false, a1, false, bOnes, (short)0, acc, false, false);
        }

        // C/D layout: VGPR r at lanes 0-15 holds M=r, lanes 16-31 hold M=r+8.
        // Every column is identical (B==1), so each lane reads its own copy.
        int base = g * 16 + half * 8;
        #pragma unroll
        for (int r = 0; r < 8; ++r) {
            int samp = base + r;
            if (samp < n) {
                unsigned long long key =
                    ((unsigned long long)__float_as_uint(acc[r]) << 32) |
                    (unsigned int)samp;
                if (key < bestKey) bestKey = key;
            }
        }
    }

    // wave32 min-reduction of packed keys (dist bits high, index low)
    #pragma unroll
    for (int off = 16; off > 0; off >>= 1) {
        unsigned int hi  = (unsigned int)(bestKey >> 32);
        unsigned int lo  = (unsigned int)(bestKey & 0xffffffffu);
        unsigned int ohi = __shfl_xor(hi, off, 32);
        unsigned int olo = __shfl_xor(lo, off, 32);
        unsigned long long other = ((unsigned long long)ohi << 32) | olo;
        if (other < bestKey) bestKey = other;
    }
    if (lane == 0) atomicMin(ws, bestKey);
}

// ---------------------------------------------------------------------------
// Kernel 3: gather target row of the winning sample
// ---------------------------------------------------------------------------
__global__ void nn_gather_kernel(const float* __restrict__ tgt,
                                 const unsigned long long* __restrict__ ws,
                                 float* __restrict__ out, int count) {
    int idx = (int)(ws[0] & 0xffffffffull);
    for (int i = blockIdx.x * blockDim.x + threadIdx.x; i < count;
         i += gridDim.x * blockDim.x) {
        out[i] = tgt[(size_t)idx * count + i];
    }
}

// ---------------------------------------------------------------------------
extern "C" void kernel_launch(void* const* d_in, const int* in_sizes, int n_in,
                              void* d_out, int out_size, void* d_ws, size_t ws_size,
                              hipStream_t stream) {
    const float* in_vel = (const float*)d_in[0];   // [1,16,66]
    const float* obs    = (const float*)d_in[1];   // [N,16,66]
    const float* tgt    = (const float*)d_in[2];   // [N,25,66]
    float* out = (float*)d_out;
    unsigned long long* ws = (unsigned long long*)d_ws;

    int n        = in_sizes[1] / DOBS;
    int n_groups = (n + 15) / 16;

    nn_init_kernel<<<1, 64, 0, stream>>>(ws);

    const int threads = 256;                 // 8 waves per block
    int blocks = (n_groups + 7) / 8;         // ~1 group per wave
    if (blocks > 1280) blocks = 1280;
    if (blocks < 1) blocks = 1;
    nn_dist_kernel<<<blocks, threads, 0, stream>>>(in_vel, obs, n, n_groups, ws);

    int cblocks = (out_size + 255) / 256;
    if (cblocks < 1) cblocks = 1;
    nn_gather_kernel<<<cblocks, 256, 0, stream>>>(tgt, ws, out, out_size);
}